// SCConv_32306744000652
// MI455X (gfx1250) — compile-verified
//
#include <hip/hip_runtime.h>

typedef __attribute__((ext_vector_type(2))) float v2f;
typedef __attribute__((ext_vector_type(8))) float v8f;

#define DIMC 64       // output feature dim
#define KDIM 128      // powcat input dim (2*64)
#define WT_STRIDE 130 // padded LDS stride (floats) -> conflict-free strided reads

// ---------------------------------------------------------------------------
// Zero the fused accumulators (d_ws is poisoned; we also accumulate, so we
// must re-zero every call).
// ---------------------------------------------------------------------------
__global__ __launch_bounds__(256) void zero_f32(float* __restrict__ p, size_t n) {
  size_t i = (size_t)blockIdx.x * blockDim.x + threadIdx.x;
  size_t stride = (size_t)gridDim.x * blockDim.x;
  for (; i < n; i += stride) p[i] = 0.0f;
}

// ---------------------------------------------------------------------------
// Fused powcat + GEMM:  G = [X, X^2] @ W + b   via V_WMMA_F32_16X16X4_F32.
//
// Block = 256 threads = 8 waves. Each wave owns one FULL 16x64 output strip:
// it loads its A tile (16 rows of X) from global exactly once and runs 4
// accumulator chains (one per 16-col N-tile), reading B from LDS. This cuts
// global A traffic 4x vs. one-tile-per-wave and gives the scheduler 4
// independent WMMA chains to cover the f32-WMMA RAW latency.
//
// Lane layouts per CDNA5 ISA 7.12.2 (f32, wave32):
//   A 16x4 : lane%16 = M row; VGPR j holds K = 2*(lane/16)+j
//   B 4x16 : lane%16 = N col; VGPR j holds K = 2*(lane/16)+j
//   C/D    : VGPR v holds element [M = v + 8*(lane/16)][N = lane%16]
// ---------------------------------------------------------------------------
__global__ __launch_bounds__(256) void gemm_powcat_wmma(
    const float* __restrict__ X,    // [nrows, 64]
    const float* __restrict__ W,    // [128, 64] row-major
    const float* __restrict__ bias, // [64]
    float* __restrict__ G,          // [nrows, 64]
    int nrows)
{
  __shared__ float Wt[DIMC * WT_STRIDE]; // Wt[n][k] = W[k][n]
  __shared__ float bs[DIMC];

  for (int idx = threadIdx.x; idx < DIMC * KDIM; idx += 256) {
    int k = idx & (KDIM - 1);
    int n = idx >> 7;
    Wt[n * WT_STRIDE + k] = W[k * DIMC + n];
  }
  if (threadIdx.x < DIMC) bs[threadIdx.x] = bias[threadIdx.x];
  __syncthreads();

  const int wave = threadIdx.x >> 5;
  const int lane = threadIdx.x & 31;
  const int half = lane >> 4;  // 0 or 1
  const int l16  = lane & 15;

  const int m0 = (blockIdx.x * 8 + wave) * 16;
  if (m0 >= nrows) return;  // wave-uniform: EXEC stays all-ones for WMMA

  int row = m0 + l16;
  if (row >= nrows) row = nrows - 1;  // tail: dup rows, stores are guarded
  const float* xrow = X + (size_t)row * DIMC;

  const float* w0 = &Wt[l16 * WT_STRIDE];   // N-tile 0 (cols  0..15)
  const float* w1 = w0 + 16 * WT_STRIDE;    // N-tile 1 (cols 16..31)
  const float* w2 = w1 + 16 * WT_STRIDE;    // N-tile 2 (cols 32..47)
  const float* w3 = w2 + 16 * WT_STRIDE;    // N-tile 3 (cols 48..63)

  v8f c0 = {}, c1 = {}, c2 = {}, c3 = {};
#pragma unroll
  for (int kk = 0; kk < KDIM; kk += 4) {
    const int ka = kk + 2 * half;
    v2f a;
    if (ka < DIMC) {
      a = *(const v2f*)(xrow + ka);           // X part of powcat
    } else {
      v2f u = *(const v2f*)(xrow + (ka - DIMC));
      a = u * u;                               // X^2 part of powcat
    }
    v2f b0 = *(const v2f*)(w0 + ka);
    v2f b1 = *(const v2f*)(w1 + ka);
    v2f b2 = *(const v2f*)(w2 + ka);
    v2f b3 = *(const v2f*)(w3 + ka);
    c0 = __builtin_amdgcn_wmma_f32_16x16x4_f32(false, a, false, b0, (short)0, c0, false, false);
    c1 = __builtin_amdgcn_wmma_f32_16x16x4_f32(false, a, false, b1, (short)0, c1, false, false);
    c2 = __builtin_amdgcn_wmma_f32_16x16x4_f32(false, a, false, b2, (short)0, c2, false, false);
    c3 = __builtin_amdgcn_wmma_f32_16x16x4_f32(false, a, false, b3, (short)0, c3, false, false);
  }

  const float bv0 = bs[ 0 + l16];
  const float bv1 = bs[16 + l16];
  const float bv2 = bs[32 + l16];
  const float bv3 = bs[48 + l16];
  float* gbase = G + (size_t)m0 * DIMC + l16;
#pragma unroll
  for (int v = 0; v < 8; ++v) {
    const int m = v + 8 * half;
    if (m0 + m < nrows) {
      float* gm = gbase + (size_t)m * DIMC;
      gm[ 0] = c0[v] + bv0;
      gm[16] = c1[v] + bv1;
      gm[32] = c2[v] + bv2;
      gm[48] = c3[v] + bv3;
    }
  }
}

// ---------------------------------------------------------------------------
// COO SpMM scatter: acc[rows[e], :] += vals[e] * G[cols[e], :]
// One wave per nnz; each lane handles a float2 of the 64-wide row.
// All three fused accumulators (141 MB) fit in the 192 MB L2, so the fp32
// atomics resolve on-chip.
// ---------------------------------------------------------------------------
__global__ __launch_bounds__(256) void spmm_scatter_atomic(
    const int* __restrict__ rows, const int* __restrict__ cols,
    const float* __restrict__ vals, const float* __restrict__ G,
    float* __restrict__ acc, int nnz)
{
  const int t = blockIdx.x * blockDim.x + threadIdx.x;
  const int e = t >> 5;
  const int lane = t & 31;
  if (e >= nnz) return;
  const int r  = rows[e];
  const int ci = cols[e];
  const float v = vals[e];
  v2f g = *(const v2f*)(G + (size_t)ci * DIMC + 2 * lane);
  float* dst = acc + (size_t)r * DIMC + 2 * lane;
  atomicAdd(dst,     v * g.x);
  atomicAdd(dst + 1, v * g.y);
}

// ---------------------------------------------------------------------------
// Y = scale * relu(acc); scale = 0.5 for Y0/Y2 segments, 1/3 for Y1.
// ---------------------------------------------------------------------------
__global__ __launch_bounds__(256) void relu_scale_out(
    const float* __restrict__ acc, float* __restrict__ out,
    size_t n0e, size_t n01e, size_t total)
{
  size_t i = (size_t)blockIdx.x * blockDim.x + threadIdx.x;
  size_t stride = (size_t)gridDim.x * blockDim.x;
  for (; i < total; i += stride) {
    float v = acc[i];
    v = v > 0.0f ? v : 0.0f;
    float s = (i < n0e || i >= n01e) ? 0.5f : (1.0f / 3.0f);
    out[i] = s * v;
  }
}

// ---------------------------------------------------------------------------
// Input order (setup_inputs dict): 0..2 X0/X1/X2; then 7 sparse triples
// (rows,cols,vals): L0@3, L1@6, L2@9, D1invB1@12, D2B1TD1inv@15,
// B2TD2inv@18, B2D3@21; then W/b pairs: n2n@24, n2e@26, e2e@28, e2n@30,
// e2t@32, t2e@34, t2t@36.
// ---------------------------------------------------------------------------
extern "C" void kernel_launch(void* const* d_in, const int* in_sizes, int n_in,
                              void* d_out, int out_size, void* d_ws, size_t ws_size,
                              hipStream_t stream) {
  (void)n_in; (void)out_size; (void)ws_size;

  const float* X0 = (const float*)d_in[0];
  const float* X1 = (const float*)d_in[1];
  const float* X2 = (const float*)d_in[2];
  const int N0 = in_sizes[0] / DIMC;
  const int N1 = in_sizes[1] / DIMC;
  const int N2 = in_sizes[2] / DIMC;

  float* acc  = (float*)d_ws;
  float* acc0 = acc;
  float* acc1 = acc + (size_t)N0 * DIMC;
  float* acc2 = acc + (size_t)(N0 + N1) * DIMC;
  float* G    = acc + (size_t)(N0 + N1 + N2) * DIMC; // reused per job

  const size_t total = (size_t)(N0 + N1 + N2) * DIMC;
  zero_f32<<<2048, 256, 0, stream>>>(acc, total);

  struct Job { const float* X; int nr; int wi; int si; float* seg; };
  const Job jobs[7] = {
    { X0, N0, 24, 3,  acc0 },  // n2n : spmm L0
    { X0, N0, 26, 15, acc1 },  // n2e : spmm D2B1TD1inv
    { X1, N1, 30, 12, acc0 },  // e2n : spmm D1invB1
    { X1, N1, 28, 6,  acc1 },  // e2e : spmm L1
    { X1, N1, 32, 18, acc2 },  // e2t : spmm B2TD2inv
    { X2, N2, 36, 9,  acc2 },  // t2t : spmm L2
    { X2, N2, 34, 21, acc1 },  // t2e : spmm B2D3
  };

  for (int j = 0; j < 7; ++j) {
    const Job& jb = jobs[j];
    const int gblocks = (jb.nr + 127) / 128; // 8 waves x 16-row strips
    gemm_powcat_wmma<<<gblocks, 256, 0, stream>>>(
        jb.X, (const float*)d_in[jb.wi], (const float*)d_in[jb.wi + 1], G, jb.nr);

    const int nnz = in_sizes[jb.si];
    const long long threads = (long long)nnz * 32;
    const int sblocks = (int)((threads + 255) / 256);
    spmm_scatter_atomic<<<sblocks, 256, 0, stream>>>(
        (const int*)d_in[jb.si], (const int*)d_in[jb.si + 1],
        (const float*)d_in[jb.si + 2], G, jb.seg, nnz);
  }

  relu_scale_out<<<2048, 256, 0, stream>>>(
      acc, (float*)d_out, (size_t)N0 * DIMC, (size_t)(N0 + N1) * DIMC, total);
}